// GNN_29008209117881
// MI455X (gfx1250) — compile-verified
//
#include <hip/hip_runtime.h>
#include <hip/hip_bf16.h>
#include <math.h>

// ---------------------------------------------------------------------------
// 2-layer GAT for MI455X (gfx1250, wave32).
// GEMMs via V_WMMA_F32_16X16X4_F32 (fp32 in/out, matches reference precision;
// GEMMs are memory-bound at 23.3 TB/s so fp32 WMMA costs nothing vs bf16).
// Dual accumulators break the WMMA D->C dependency chain (2x WMMA ILP).
// Edge aggregation via device-built CSR-by-dst: one block (8 waves = 8 heads)
// per destination node, softmax max/sum held in registers with wave32
// __shfl_xor reductions -> zero output atomics on the 218M-MAC scatter phase.
// ---------------------------------------------------------------------------

typedef __attribute__((ext_vector_type(2))) float v2f;
typedef __attribute__((ext_vector_type(8))) float v8f;

#define HEADS 8
#define D_IN 256
#define NEG_SLOPE 0.2f

// ---------------- GEMM: OUT[M,NC] = X[M,K] @ W[K,NC] via WMMA f32 -----------
// One wave computes one 16x16 tile; 8 waves/block cover 8 column tiles.
template<int K, int NC>
__global__ __launch_bounds__(256) void gemm_wmma_f32(
    const float* __restrict__ X, const float* __restrict__ W,
    float* __restrict__ OUT, int M)
{
  const int lane = threadIdx.x & 31;
  const int wave = threadIdx.x >> 5;
  const int half = lane >> 4;        // 0: K lo pair / M 0-7 ; 1: K hi pair / M 8-15
  const int l16  = lane & 15;
  const int row0 = blockIdx.x * 16;
  const int col0 = (blockIdx.y * 8 + wave) * 16;

  int rowA = row0 + l16;
  if (rowA >= M) rowA = M - 1;       // clamp (stores are guarded below)
  const float* xrow = X + (size_t)rowA * K;
  const float* wcol = W + col0 + l16;

  // Two independent accumulator chains -> 2x v_wmma ILP per wave.
  v8f c0 = {}, c1 = {};
#pragma unroll 2
  for (int k = 0; k < K; k += 8) {
    // A 16x4 f32 layout: lanes 0-15 hold K={k,k+1}, lanes 16-31 K={k+2,k+3}
    v2f a0 = *(const v2f*)(xrow + k + 2 * half);
    v2f a1 = *(const v2f*)(xrow + k + 4 + 2 * half);
    // B 4x16 f32 layout (row striped across lanes)
    v2f b0, b1;
    b0.x = wcol[(size_t)(k + 2 * half    ) * NC];
    b0.y = wcol[(size_t)(k + 2 * half + 1) * NC];
    b1.x = wcol[(size_t)(k + 4 + 2 * half    ) * NC];
    b1.y = wcol[(size_t)(k + 4 + 2 * half + 1) * NC];
    c0 = __builtin_amdgcn_wmma_f32_16x16x4_f32(false, a0, false, b0,
                                               (short)0, c0, false, false);
    c1 = __builtin_amdgcn_wmma_f32_16x16x4_f32(false, a1, false, b1,
                                               (short)0, c1, false, false);
  }

  // C/D layout: VGPR r -> lanes 0-15: (M=r, N=lane), lanes 16-31: (M=r+8, N=lane-16)
  if (row0 + 16 <= M) {
    // fast path: full tile, unguarded stores (no exec-mask churn)
    float* obase = OUT + (size_t)(row0 + half * 8) * NC + col0 + l16;
#pragma unroll
    for (int r = 0; r < 8; ++r) obase[(size_t)r * NC] = c0[r] + c1[r];
  } else {
#pragma unroll
    for (int r = 0; r < 8; ++r) {
      int row = row0 + r + half * 8;
      if (row < M) OUT[(size_t)row * NC + col0 + l16] = c0[r] + c1[r];
    }
  }
}

// ------------- per-node attention coefficients: a = <h[n,h,:], att[h,:]> ----
__global__ void attn_coef_kernel(const float* __restrict__ Hf,
                                 const float* __restrict__ att_s,
                                 const float* __restrict__ att_d,
                                 float* __restrict__ a_s, float* __restrict__ a_d,
                                 int N, int C)
{
  int idx = blockIdx.x * blockDim.x + threadIdx.x;   // one (node, head) pair
  if (idx >= N * HEADS) return;
  int n = idx >> 3, h = idx & 7;
  const float* row = Hf + (size_t)n * HEADS * C + h * C;
  float ss = 0.f, sd = 0.f;
  for (int cc = 0; cc < C; ++cc) {
    float v = row[cc];
    ss += v * att_s[h * C + cc];
    sd += v * att_d[h * C + cc];
  }
  a_s[idx] = ss;
  a_d[idx] = sd;
}

// ---------------- CSR construction (dst-sorted edge list) -------------------
__global__ void degree_count_kernel(const long long* __restrict__ ei,
                                    int E, int N, int* __restrict__ deg)
{
  int idx = blockIdx.x * blockDim.x + threadIdx.x;
  if (idx >= E + N) return;
  int dst = (idx < E) ? (int)ei[(size_t)E + idx] : (idx - E);  // self-loops appended
  atomicAdd(&deg[dst], 1);
}

__global__ void scan_excl_kernel(const int* __restrict__ deg,
                                 int* __restrict__ rowptr,
                                 int* __restrict__ cursor, int N)
{
  __shared__ int buf[1024];
  __shared__ int carry;
  int tid = threadIdx.x;
  if (tid == 0) carry = 0;
  __syncthreads();
  for (int base = 0; base < N; base += 1024) {
    int i = base + tid;
    int v = (i < N) ? deg[i] : 0;
    buf[tid] = v;
    __syncthreads();
    for (int off = 1; off < 1024; off <<= 1) {
      int t = (tid >= off) ? buf[tid - off] : 0;
      __syncthreads();
      buf[tid] += t;
      __syncthreads();
    }
    int excl = buf[tid] - v + carry;
    if (i < N) { rowptr[i] = excl; cursor[i] = excl; }
    __syncthreads();
    if (tid == 1023) carry += buf[1023];
    __syncthreads();
  }
  if (tid == 0) rowptr[N] = carry;
}

__global__ void scatter_edges_kernel(const long long* __restrict__ ei,
                                     int E, int N, int* __restrict__ cursor,
                                     int* __restrict__ csr_src,
                                     int* __restrict__ csr_eid)
{
  int idx = blockIdx.x * blockDim.x + threadIdx.x;
  if (idx >= E + N) return;
  int src, dst;
  if (idx < E) { src = (int)ei[idx]; dst = (int)ei[(size_t)E + idx]; }
  else         { src = idx - E;      dst = idx - E; }
  int pos = atomicAdd(&cursor[dst], 1);
  csr_src[pos] = src;
  csr_eid[pos] = idx;
}

// ---------------- GAT softmax-aggregation: one block per dst node -----------
// 8 waves/block; wave h owns head h. wave32 shfl reductions for max & sum.
template<int C, bool ELU, bool WRITE_ALPHA>
__global__ __launch_bounds__(256) void gat_aggregate_kernel(
    const float* __restrict__ Hf,      // [N, H*C] pre-aggregation features
    const float* __restrict__ a_src,   // [N, H]
    const float* __restrict__ a_dst,   // [N, H]
    const int* __restrict__ rowptr,
    const int* __restrict__ csr_src,
    const int* __restrict__ csr_eid,
    const float* __restrict__ bias,    // [H*C]
    float* __restrict__ Out,           // [N, H*C]
    float* __restrict__ alpha_out)     // [(E+N)*H] or unused
{
  const int node = blockIdx.x;
  const int lane = threadIdx.x & 31;
  const int head = threadIdx.x >> 5;
  const int start = rowptr[node];
  const int end   = rowptr[node + 1];
  const float adst = a_dst[node * HEADS + head];

  // pass A: segment max of leakyrelu logits (register reduction, no atomics)
  float m = -INFINITY;
  for (int j = start + lane; j < end; j += 32) {
    float z = a_src[csr_src[j] * HEADS + head] + adst;
    z = (z > 0.f) ? z : NEG_SLOPE * z;
    m = fmaxf(m, z);
  }
#pragma unroll
  for (int off = 16; off; off >>= 1) m = fmaxf(m, __shfl_xor(m, off, 32));
  if (m == -INFINITY) m = 0.f;       // reference: where(isfinite(m), m, 0)

  // pass B: segment sum of exp(z - m)   (a_src rows are L2-resident: 1.6 MB)
  float s = 0.f;
  for (int j = start + lane; j < end; j += 32) {
    float z = a_src[csr_src[j] * HEADS + head] + adst;
    z = (z > 0.f) ? z : NEG_SLOPE * z;
    s += __expf(z - m);
  }
#pragma unroll
  for (int off = 16; off; off >>= 1) s += __shfl_xor(s, off, 32);
  const float inv_denom = 1.f / (s + 1e-16f);

  // pass C: out[dst,h,:] += alpha * h[src,h,:]; lane = channel accumulator
  float acc = 0.f;
  for (int j = start; j < end; ++j) {
    int srcn = csr_src[j];
    float z = a_src[srcn * HEADS + head] + adst;   // broadcast load (same addr all lanes)
    z = (z > 0.f) ? z : NEG_SLOPE * z;
    float alpha = __expf(z - m) * inv_denom;
    if (WRITE_ALPHA && lane == 0)
      alpha_out[(size_t)csr_eid[j] * HEADS + head] = alpha;
    if (lane < C)
      acc += alpha * Hf[(size_t)srcn * (HEADS * C) + head * C + lane];
  }
  if (lane < C) {
    float v = acc + bias[head * C + lane];
    if (ELU) v = (v > 0.f) ? v : expm1f(v);
    Out[(size_t)node * (HEADS * C) + head * C + lane] = v;
  }
}

// ---------------------------------------------------------------------------
extern "C" void kernel_launch(void* const* d_in, const int* in_sizes, int n_in,
                              void* d_out, int out_size, void* d_ws, size_t ws_size,
                              hipStream_t stream)
{
  const float*     x    = (const float*)d_in[0];
  const long long* ei   = (const long long*)d_in[1];   // int64 [2, E]
  // d_in[2] = edge_attr: ignored (GATConv without edge_dim)
  const float*     W1   = (const float*)d_in[3];
  const float*     as1w = (const float*)d_in[4];
  const float*     ad1w = (const float*)d_in[5];
  const float*     b1   = (const float*)d_in[6];
  const float*     W2   = (const float*)d_in[7];
  const float*     as2w = (const float*)d_in[8];
  const float*     ad2w = (const float*)d_in[9];
  const float*     b2   = (const float*)d_in[10];

  const int N  = in_sizes[0] / D_IN;   // 50000
  const int E  = in_sizes[1] / 2;      // 800000
  const int ET = E + N;                // edges incl. self-loops

  // ---- workspace carve-out (256B aligned regions) ----
  char* p = (char*)d_ws;
  auto alloc = [&](size_t bytes) -> void* {
    void* r = (void*)p;
    p += (bytes + 255) & ~(size_t)255;
    return r;
  };
  float* h1     = (float*)alloc((size_t)N * 256 * sizeof(float)); // x @ W1
  float* o1     = (float*)alloc((size_t)N * 256 * sizeof(float)); // layer1 out (post ELU)
  float* h2     = (float*)alloc((size_t)N * 128 * sizeof(float)); // o1 @ W2
  float* as1    = (float*)alloc((size_t)N * HEADS * sizeof(float));
  float* ad1    = (float*)alloc((size_t)N * HEADS * sizeof(float));
  float* as2    = (float*)alloc((size_t)N * HEADS * sizeof(float));
  float* ad2    = (float*)alloc((size_t)N * HEADS * sizeof(float));
  int*   deg    = (int*)alloc((size_t)N * sizeof(int));
  int*   rowptr = (int*)alloc((size_t)(N + 1) * sizeof(int));
  int*   cursor = (int*)alloc((size_t)N * sizeof(int));
  int*   csr_src= (int*)alloc((size_t)ET * sizeof(int));
  int*   csr_eid= (int*)alloc((size_t)ET * sizeof(int));
  (void)ws_size;

  // ---- build CSR by destination ----
  hipMemsetAsync(deg, 0, (size_t)N * sizeof(int), stream);
  degree_count_kernel<<<(ET + 255) / 256, 256, 0, stream>>>(ei, E, N, deg);
  scan_excl_kernel<<<1, 1024, 0, stream>>>(deg, rowptr, cursor, N);
  scatter_edges_kernel<<<(ET + 255) / 256, 256, 0, stream>>>(ei, E, N, cursor,
                                                             csr_src, csr_eid);

  // ---- layer 1: GEMM (WMMA f32) + coefficients + aggregate(+bias+ELU) ----
  dim3 g1((N + 15) / 16, 2);           // 16 col tiles / (8 waves) = grid.y 2
  gemm_wmma_f32<256, 256><<<g1, 256, 0, stream>>>(x, W1, h1, N);
  attn_coef_kernel<<<(N * HEADS + 255) / 256, 256, 0, stream>>>(h1, as1w, ad1w,
                                                                as1, ad1, N, 32);
  gat_aggregate_kernel<32, true, false><<<N, 256, 0, stream>>>(
      h1, as1, ad1, rowptr, csr_src, csr_eid, b1, o1, nullptr);

  // ---- layer 2: GEMM (WMMA f32) + coefficients + aggregate(+bias, alpha) ----
  dim3 g2((N + 15) / 16, 1);           // 8 col tiles = 8 waves
  gemm_wmma_f32<256, 128><<<g2, 256, 0, stream>>>(o1, W2, h2, N);
  attn_coef_kernel<<<(N * HEADS + 255) / 256, 256, 0, stream>>>(h2, as2w, ad2w,
                                                                as2, ad2, N, 16);
  float* out_h2    = (float*)d_out;                 // [N, 128]
  float* out_alpha = out_h2 + (size_t)N * 128;      // [(E+N), 8]
  gat_aggregate_kernel<16, false, true><<<N, 256, 0, stream>>>(
      h2, as2, ad2, rowptr, csr_src, csr_eid, b2, out_h2, out_alpha);
}